// YoloLayer_30545807409246
// MI455X (gfx1250) — compile-verified
//
#include <hip/hip_runtime.h>
#include <stdint.h>

// Problem constants (from reference)
constexpr int kNB = 16;
constexpr int kNA = 3;
constexpr int kNC = 80;
constexpr int kNH = 76;
constexpr int kNW = 76;
constexpr int kMAXT = 50;
constexpr int kS = kNH * kNW;              // 5776 (channel stride in floats)
constexpr int kNCELL = kNA * kS;           // 17328 cells per batch
constexpr int kNTOT = kNB * kNCELL;        // 277248 total cells
constexpr int kCPB = 256;                  // cells per block
constexpr int kBPB = (kNCELL + kCPB - 1) / kCPB;   // 68 blocks per batch
constexpr int kNBLK = kNB * kBPB;          // 1088 main blocks

// anchors[ANCHOR_MASK] / STRIDE
__device__ __constant__ float c_aw[3] = {10.f/8.f, 16.f/8.f, 33.f/8.f};
__device__ __constant__ float c_ah[3] = {13.f/8.f, 30.f/8.f, 23.f/8.f};

// ---------------------------------------------------------------------------
// Main loss kernel (first in module so its disasm shows in the snippet).
// Stages this batch's 50 precomputed GT records into LDS with the CDNA5
// async global->LDS path (ASYNCcnt), then computes conf/coord/cls loss terms.
// ---------------------------------------------------------------------------
__global__ __launch_bounds__(256) void k_main(const float* __restrict__ out,
                                              const float* __restrict__ hit,
                                              const float* __restrict__ tcoord,
                                              const float* __restrict__ tcls,
                                              const float* __restrict__ gt,
                                              float* __restrict__ partials) {
    __shared__ float4 sgt[2 * kMAXT];   // 50 GT x {x1,x2,y1,y2 | area,0,0,0}
    __shared__ float  sred[256];

    const int blk = blockIdx.x;
    const int b = blk / kBPB;
    const int cell0 = (blk % kBPB) * kCPB;

    // CDNA5 async global->LDS stage (tracked by ASYNCcnt, not LOADcnt).
    if (threadIdx.x < 2 * kMAXT) {
        unsigned lds_off = (unsigned)(uintptr_t)(&sgt[threadIdx.x]);
        uint64_t ga = (uint64_t)(uintptr_t)(gt + (size_t)b * kMAXT * 8
                                               + (size_t)threadIdx.x * 4);
        asm volatile("global_load_async_to_lds_b128 %0, %1, off"
                     :: "v"(lds_off), "v"(ga) : "memory");
    }
    asm volatile("s_wait_asynccnt 0x0" ::: "memory");
    __syncthreads();

    float acc = 0.0f;
    const int cell = cell0 + threadIdx.x;
    if (cell < kNCELL) {
        const int a = cell / kS;
        const int r = cell % kS;
        const int j = r / kNW;
        const int i = r % kNW;
        const float* base = out + ((size_t)b * kNA + a) * (5 + kNC) * kS + r;

        const float o0 = base[0];
        const float o1 = base[(size_t)1 * kS];
        const float o2 = base[(size_t)2 * kS];
        const float o3 = base[(size_t)3 * kS];
        const float o4 = base[(size_t)4 * kS];

        const float tx = 1.0f / (1.0f + __expf(-o0));
        const float ty = 1.0f / (1.0f + __expf(-o1));
        const float px = tx + (float)i;
        const float py = ty + (float)j;
        const float pw = __expf(o2) * c_aw[a];
        const float ph = __expf(o3) * c_ah[a];
        const float px1 = px - 0.5f * pw, px2 = px + 0.5f * pw;
        const float py1 = py - 0.5f * ph, py2 = py + 0.5f * ph;
        const float parea = pw * ph;

        // max_t IoU <= 0.5  <=>  for all t: 3*inter <= areaP + areaG (div-free;
        // union >= areaP > 0 so the 1e-9 guard in the reference is inert)
        bool below = true;
        #pragma unroll 5
        for (int t = 0; t < kMAXT; ++t) {
            float4 bx = sgt[2*t];
            float4 ax = sgt[2*t + 1];
            float iw = fminf(px2, bx.y) - fmaxf(px1, bx.x);
            float ih = fminf(py2, bx.w) - fmaxf(py1, bx.z);
            iw = fmaxf(iw, 0.0f);
            ih = fmaxf(ih, 0.0f);
            below = below && (3.0f * (iw * ih) <= parea + ax.x);
        }

        const int flat = b * kNCELL + cell;
        const float h = hit[flat];
        const float cm = (h > 0.0f) ? 1.0f : (below ? 1.0f : 0.0f);
        const float conf = 1.0f / (1.0f + __expf(-o4));
        const float tconf = (h > 0.0f) ? 1.0f : 0.0f;
        const float dconf = conf * cm - tconf * cm;
        acc = dconf * dconf;                          // loss_conf term

        if (h > 0.0f) {
            // coord loss (coord_mask == 1 here)
            const float c0 = tx - tcoord[(size_t)flat*4 + 0];
            const float c1 = ty - tcoord[(size_t)flat*4 + 1];
            const float c2 = o2 - tcoord[(size_t)flat*4 + 2];
            const float c3 = o3 - tcoord[(size_t)flat*4 + 3];
            acc += 0.5f * (c0*c0 + c1*c1 + c2*c2 + c3*c3);
            // class CE (log-softmax over 80 strided logits, hit cells only)
            int tc = (int)tcls[flat];
            tc = tc < 0 ? 0 : (tc > kNC - 1 ? kNC - 1 : tc);
            const float* cl = base + (size_t)5 * kS;
            float mx = -1e30f;
            for (int c = 0; c < kNC; ++c) mx = fmaxf(mx, cl[(size_t)c * kS]);
            float sum = 0.0f;
            for (int c = 0; c < kNC; ++c) sum += __expf(cl[(size_t)c * kS] - mx);
            acc += (mx + __logf(sum)) - cl[(size_t)tc * kS];
        }
    }

    sred[threadIdx.x] = acc;
    __syncthreads();
    for (int s = 128; s > 0; s >>= 1) {
        if (threadIdx.x < (unsigned)s) sred[threadIdx.x] += sred[threadIdx.x + s];
        __syncthreads();
    }
    if (threadIdx.x == 0) partials[blk] = sred[0];
}

// Zero-fill scatter arrays, 16B per lane.
__global__ void k_zero4(float4* __restrict__ p, int n4) {
    int i = blockIdx.x * blockDim.x + threadIdx.x;
    if (i < n4) p[i] = make_float4(0.f, 0.f, 0.f, 0.f);
}

// One lane per batch element: compute GT records (corners + area, zeroed when
// invalid per the cumprod-validity rule) and do the serialized scatter of
// hit / tcoord / tcls (last-write-wins, deterministic).
__global__ void k_prep(const float* __restrict__ target,
                       float* __restrict__ hit,
                       float* __restrict__ tcoord,
                       float* __restrict__ tcls,
                       float* __restrict__ gt) {
    int b = threadIdx.x;
    if (b >= kNB) return;
    const float* tb = target + (size_t)b * kMAXT * 5;
    float* g = gt + (size_t)b * kMAXT * 8;
    bool valid = true;
    for (int t = 0; t < kMAXT; ++t) {
        float cls = tb[t*5 + 0];
        float x = tb[t*5 + 1], y = tb[t*5 + 2];
        float w = tb[t*5 + 3], h = tb[t*5 + 4];
        valid = valid && (x != 0.0f);               // cumprod semantics
        float gx = x * (float)kNW, gy = y * (float)kNH;
        float gw = w * 76.0f,      gh = h * 76.0f;  // NET/STRIDE = 76
        if (!valid) { gx = 0.f; gy = 0.f; gw = 0.f; gh = 0.f; }
        g[t*8 + 0] = gx - 0.5f * gw;   // x1
        g[t*8 + 1] = gx + 0.5f * gw;   // x2
        g[t*8 + 2] = gy - 0.5f * gh;   // y1
        g[t*8 + 3] = gy + 0.5f * gh;   // y2
        g[t*8 + 4] = gw * gh;          // area
        g[t*8 + 5] = 0.f; g[t*8 + 6] = 0.f; g[t*8 + 7] = 0.f;
        if (valid) {
            int bn = 0; float best = -1.0f;
            for (int a = 0; a < 3; ++a) {
                float inter = fminf(gw, c_aw[a]) * fminf(gh, c_ah[a]);
                float u = gw * gh + c_aw[a] * c_ah[a] - inter;
                float wi = inter / fmaxf(u, 1e-9f);
                if (wi > best) { best = wi; bn = a; }   // argmax: first max wins
            }
            int gi = (int)gx, gj = (int)gy;
            int flat = ((b * kNA + bn) * kNH + gj) * kNW + gi;
            hit[flat] = 1.0f;
            tcoord[(size_t)flat*4 + 0] = gx - (float)gi;
            tcoord[(size_t)flat*4 + 1] = gy - (float)gj;
            tcoord[(size_t)flat*4 + 2] = __logf(fmaxf(gw / c_aw[bn], 1e-9f));
            tcoord[(size_t)flat*4 + 3] = __logf(fmaxf(gh / c_ah[bn], 1e-9f));
            tcls[flat] = cls;
        }
    }
}

// Fixed-order final reduction: deterministic, overwrites d_out poison.
__global__ void k_reduce(const float* __restrict__ partials, int n,
                         float* __restrict__ outp) {
    __shared__ float s[256];
    float acc = 0.0f;
    for (int i = threadIdx.x; i < n; i += 256) acc += partials[i];
    s[threadIdx.x] = acc;
    __syncthreads();
    for (int st = 128; st > 0; st >>= 1) {
        if (threadIdx.x < (unsigned)st) s[threadIdx.x] += s[threadIdx.x + st];
        __syncthreads();
    }
    if (threadIdx.x == 0) outp[0] = s[0];
}

extern "C" void kernel_launch(void* const* d_in, const int* in_sizes, int n_in,
                              void* d_out, int out_size, void* d_ws, size_t ws_size,
                              hipStream_t stream) {
    (void)in_sizes; (void)n_in; (void)out_size; (void)ws_size;
    const float* output = (const float*)d_in[0];   // (16, 255, 76, 76) fp32
    const float* target = (const float*)d_in[1];   // (16, 250) fp32

    float* ws       = (float*)d_ws;
    float* hit      = ws;                          // [NTOT]
    float* tcoord   = ws + (size_t)kNTOT;          // [NTOT*4]
    float* tcls     = ws + (size_t)5 * kNTOT;      // [NTOT]
    float* gtrec    = ws + (size_t)6 * kNTOT;      // [NB*MAXT*8] (16B-aligned)
    float* partials = gtrec + (size_t)kNB * kMAXT * 8;  // [kNBLK]

    const int nz4 = (6 * kNTOT) / 4;               // zero scatter arrays (float4)
    k_zero4<<<(nz4 + 255) / 256, 256, 0, stream>>>((float4*)ws, nz4);
    k_prep<<<1, 32, 0, stream>>>(target, hit, tcoord, tcls, gtrec);
    k_main<<<kNBLK, 256, 0, stream>>>(output, hit, tcoord, tcls, gtrec, partials);
    k_reduce<<<1, 256, 0, stream>>>(partials, kNBLK, (float*)d_out);
}